// GCN2_13460427506085
// MI455X (gfx1250) — compile-verified
//
#include <hip/hip_runtime.h>

typedef __attribute__((ext_vector_type(2))) float v2f;
typedef __attribute__((ext_vector_type(8))) float v8f;

// ---------------------------------------------------------------------------
// Degree count: one thread per edge, atomic +1.0 at dst.
// ---------------------------------------------------------------------------
__global__ void k_deg(const long long* __restrict__ dst, float* __restrict__ deg, int nE) {
    int e = blockIdx.x * blockDim.x + threadIdx.x;
    if (e < nE) atomicAdd(&deg[(int)dst[e]], 1.0f);
}

// dinv = deg > 0 ? rsqrt(deg) : 0
__global__ void k_dinv(const float* __restrict__ deg, float* __restrict__ dinv, int nN) {
    int i = blockIdx.x * blockDim.x + threadIdx.x;
    if (i < nN) {
        float d = deg[i];
        dinv[i] = (d > 0.0f) ? rsqrtf(d) : 0.0f;
    }
}

// Bt[n*K + k] = B[k*N + n]  (tiny one-shot weight transpose)
template <int N, int K>
__global__ void k_transpose(const float* __restrict__ B, float* __restrict__ Bt) {
    int i = blockIdx.x * blockDim.x + threadIdx.x;   // i = n*K + k
    if (i < N * K) {
        int n = i / K, k = i % K;
        Bt[i] = B[k * N + n];
    }
}

// ---------------------------------------------------------------------------
// fp32 WMMA GEMM: C[M x N] = A[M x K] * B[K x N], exact fp32.
// B is pre-transposed (Bt[N][K]) so both A and B fetch as contiguous b64.
// One wave per 16x32 output tile (two accumulators) -> each A load feeds two
// V_WMMA_F32_16X16X4_F32 ops. K stepped by 4.
// A frag (16x4): lanes 0-15 M=lane, K=0/1 in VGPR0/1; lanes 16-31 K=2/3.
// B frag (4x16): VGPR0 K=khalf+0, VGPR1 K=khalf+1, N=lane&15 per half-wave.
// C/D: 8 VGPRs, lanes 0-15 -> M=0..7, lanes 16-31 -> M=8..15, N=lane&15.
// ---------------------------------------------------------------------------
template <int N, int K>
__global__ __launch_bounds__(128) void k_gemm_wmma(const float* __restrict__ A,
                                                   const float* __restrict__ Bt,
                                                   float* __restrict__ C, int M) {
    const int lane = threadIdx.x & 31;
    const int wave = blockIdx.x * (blockDim.x >> 5) + (threadIdx.x >> 5);
    const int NT2  = N / 32;                        // column-pair tiles
    const int mt   = wave / NT2;
    const int nt   = wave % NT2;
    if (mt * 16 >= M) return;                       // wave-uniform: EXEC stays all-1s

    int row = mt * 16 + (lane & 15);
    if (row >= M) row = M - 1;                      // clamp tail loads (stores guarded)
    const int col0  = nt * 32 + (lane & 15);
    const int khalf = (lane >> 4) << 1;             // 0 or 2

    const float* ap  = A  + (size_t)row  * K + khalf;
    const float* bp0 = Bt + (size_t)col0 * K + khalf;
    const float* bp1 = bp0 + (size_t)16  * K;

    v8f acc0 = {}, acc1 = {};
#pragma unroll 8
    for (int k = 0; k < K; k += 4) {
        v2f a  = *(const v2f*)(ap  + k);
        v2f b0 = *(const v2f*)(bp0 + k);
        v2f b1 = *(const v2f*)(bp1 + k);
        acc0 = __builtin_amdgcn_wmma_f32_16x16x4_f32(false, a, false, b0,
                                                     (short)0, acc0, false, false);
        acc1 = __builtin_amdgcn_wmma_f32_16x16x4_f32(false, a, false, b1,
                                                     (short)0, acc1, false, false);
    }

    const int mbase = mt * 16 + ((lane >> 4) << 3);
    float* cp = C + (size_t)mbase * N + nt * 32 + (lane & 15);
    if (mbase + 8 <= M) {                           // full tile: straight-line stores
#pragma unroll
        for (int v = 0; v < 8; ++v) {
            cp[(size_t)v * N]      = acc0[v];
            cp[(size_t)v * N + 16] = acc1[v];
        }
    } else {
#pragma unroll
        for (int v = 0; v < 8; ++v) {
            if (mbase + v < M) {
                cp[(size_t)v * N]      = acc0[v];
                cp[(size_t)v * N + 16] = acc1[v];
            }
        }
    }
}

// ---------------------------------------------------------------------------
// Edge scatter: one wave32 per edge. Each lane gathers VW contiguous floats
// (128-bit load for VW=4), scales by norm, atomic-adds into out[dst].
// NF = 32 * VW features per node.
// ---------------------------------------------------------------------------
template <int NF, int VW>
__global__ void k_scatter(const long long* __restrict__ src,
                          const long long* __restrict__ dst,
                          const float* __restrict__ dinv,
                          const float* __restrict__ h,
                          float* __restrict__ out, int nE) {
    const int lane = threadIdx.x & 31;
    const int e    = blockIdx.x * (blockDim.x >> 5) + (threadIdx.x >> 5);
    if (e >= nE) return;

    const int s = (int)src[e];
    const int d = (int)dst[e];
    const float norm = dinv[s] * dinv[d];

    const float* hp = h + (size_t)s * NF + lane * VW;
    float* op       = out + (size_t)d * NF + lane * VW;

    if constexpr (VW == 4) {
        float4 v = *(const float4*)hp;
        atomicAdd(op + 0, v.x * norm);
        atomicAdd(op + 1, v.y * norm);
        atomicAdd(op + 2, v.z * norm);
        atomicAdd(op + 3, v.w * norm);
    } else {
        float2 v = *(const float2*)hp;
        atomicAdd(op + 0, v.x * norm);
        atomicAdd(op + 1, v.y * norm);
    }
}

// out[i] = relu(agg[i] + bias[i % F])   (F power of two)
__global__ void k_bias_relu(const float* __restrict__ agg, const float* __restrict__ bias,
                            float* __restrict__ out, int total, int F) {
    int i = blockIdx.x * blockDim.x + threadIdx.x;
    if (i < total) {
        float v = agg[i] + bias[i & (F - 1)];
        out[i] = v > 0.0f ? v : 0.0f;
    }
}

// out[i] += bias[i % F]
__global__ void k_bias(float* __restrict__ out, const float* __restrict__ bias,
                       int total, int F) {
    int i = blockIdx.x * blockDim.x + threadIdx.x;
    if (i < total) out[i] += bias[i & (F - 1)];
}

// ---------------------------------------------------------------------------
// Launch
// inputs: x[f32 N*128], edge_index[i64 2*E], W1[f32 128*128], b1[f32 128],
//         W2[f32 128*64], b2[f32 64];  out: f32 N*64
// ---------------------------------------------------------------------------
extern "C" void kernel_launch(void* const* d_in, const int* in_sizes, int n_in,
                              void* d_out, int out_size, void* d_ws, size_t ws_size,
                              hipStream_t stream) {
    (void)n_in; (void)out_size; (void)ws_size;

    const float*     x    = (const float*)d_in[0];
    const long long* eidx = (const long long*)d_in[1];
    const float*     W1   = (const float*)d_in[2];
    const float*     b1   = (const float*)d_in[3];
    const float*     W2   = (const float*)d_in[4];
    const float*     b2   = (const float*)d_in[5];
    float*           out  = (float*)d_out;

    const int nN = in_sizes[0] / 128;
    const int nE = in_sizes[1] / 2;
    const long long* src = eidx;
    const long long* dst = eidx + nE;

    // workspace carve-out (256B aligned)
    char*  ws  = (char*)d_ws;
    size_t off = 0;
    auto alloc = [&](size_t bytes) -> void* {
        void* p = ws + off;
        off += (bytes + 255) & ~(size_t)255;
        return p;
    };
    float* deg  = (float*)alloc((size_t)nN * 4);
    float* dinv = (float*)alloc((size_t)nN * 4);
    float* W1t  = (float*)alloc((size_t)128 * 128 * 4);  // W1 transposed [N][K]
    float* W2t  = (float*)alloc((size_t)64 * 128 * 4);   // W2 transposed [N][K]
    float* h    = (float*)alloc((size_t)nN * 128 * 4);   // x @ W1, then messages
    float* agg1 = (float*)alloc((size_t)nN * 128 * 4);   // layer-1 aggregation -> h1
    float* h2   = (float*)alloc((size_t)nN * 64 * 4);    // h1 @ W2

    const int MT = (nN + 15) / 16;

    // --- normalization (shared by both layers) + weight transposes ---
    hipMemsetAsync(deg, 0, (size_t)nN * 4, stream);
    k_deg <<<(nE + 255) / 256, 256, 0, stream>>>(dst, deg, nE);
    k_dinv<<<(nN + 255) / 256, 256, 0, stream>>>(deg, dinv, nN);
    k_transpose<128, 128><<<(128 * 128 + 255) / 256, 256, 0, stream>>>(W1, W1t);
    k_transpose<64, 128><<<(64 * 128 + 255) / 256, 256, 0, stream>>>(W2, W2t);

    // --- layer 1 ---
    {   // h = x @ W1   (N=128, K=128): 4 column-pair tiles per row tile
        int waves = MT * (128 / 32);
        k_gemm_wmma<128, 128><<<(waves + 3) / 4, 128, 0, stream>>>(x, W1t, h, nN);
    }
    hipMemsetAsync(agg1, 0, (size_t)nN * 128 * 4, stream);
    k_scatter<128, 4><<<(nE + 7) / 8, 256, 0, stream>>>(src, dst, dinv, h, agg1, nE);
    k_bias_relu<<<((size_t)nN * 128 + 255) / 256, 256, 0, stream>>>(agg1, b1, agg1,
                                                                    nN * 128, 128);

    // --- layer 2 ---
    {   // h2 = h1 @ W2   (N=64, K=128)
        int waves = MT * (64 / 32);
        k_gemm_wmma<64, 128><<<(waves + 3) / 4, 128, 0, stream>>>(agg1, W2t, h2, nN);
    }
    hipMemsetAsync(out, 0, (size_t)nN * 64 * 4, stream);
    k_scatter<64, 2><<<(nE + 7) / 8, 256, 0, stream>>>(src, dst, dinv, h2, out, nE);
    k_bias<<<((size_t)nN * 64 + 255) / 256, 256, 0, stream>>>(out, b2, nN * 64, 64);
}